// SAGE_8899172237857
// MI455X (gfx1250) — compile-verified
//
#include <hip/hip_runtime.h>

#define N_NODES 100000
#define N_EDGES 1600000
#define IN_F 128
#define HID  128
#define OUT_C 64

typedef __attribute__((ext_vector_type(2))) float v2f;
typedef __attribute__((ext_vector_type(8))) float v8f;

__global__ void zero_f32(float* __restrict__ p, int n) {
    int i = blockIdx.x * blockDim.x + threadIdx.x;
    if (i < n) p[i] = 0.0f;
}

__global__ void deg_kernel(const int* __restrict__ dst, float* __restrict__ deg) {
    int e = blockIdx.x * blockDim.x + threadIdx.x;
    if (e < N_EDGES) atomicAdd(&deg[dst[e]], 1.0f);
}

__global__ void invdeg_kernel(float* __restrict__ deg) {
    int i = blockIdx.x * blockDim.x + threadIdx.x;
    if (i < N_NODES) deg[i] = 1.0f / fmaxf(deg[i], 1.0f);
}

// 2 edges per 256-thread block, 128 features each. Coalesced 128-float gather,
// f32 atomic scatter (agg is L2-resident: 51MB << 192MB L2). The mean-normalizer
// inv_deg[dst] is folded in here (constant per destination row, so it commutes
// with the segment sum); the mul hides under the atomic's memory latency and
// keeps the GEMM inner loop free of per-row scaling.
__global__ void scatter_kernel(const float* __restrict__ h,
                               const int* __restrict__ src,
                               const int* __restrict__ dst,
                               const float* __restrict__ inv_deg,
                               float* __restrict__ agg) {
    int e = blockIdx.x * 2 + (threadIdx.x >> 7);
    int f = threadIdx.x & 127;
    int s = src[e];
    int d = dst[e];
    atomicAdd(&agg[d * 128 + f], h[s * 128 + f] * inv_deg[d]);
}

// Fused dual GEMM: out = A_self @ Wself + A_neigh @ Wneigh + bias.
// K = 128 fixed. One wave -> 32 rows (two 16-row M-tiles) x NCOLS via
// V_WMMA_F32_16X16X4_F32; the two M-tiles share every B-tile load.
// __launch_bounds__(128,1): 4 waves/block, min 1 wave/SIMD -> full VGPR
// budget for the 128 accumulator VGPRs (layer 1), no scratch spills.
template<int NCOLS, bool RELU1>
__global__ void __launch_bounds__(128, 1)
sage_gemm(const float* __restrict__ A_self,   // [N_NODES,128]
          const float* __restrict__ A_neigh,  // [N_NODES,128] pre-scaled agg
          const float* __restrict__ Wself,    // [128,NCOLS]
          const float* __restrict__ Wneigh,   // [128,NCOLS]
          const float* __restrict__ bias,     // [NCOLS]
          float* __restrict__ out0,           // raw result
          float* __restrict__ out1)           // relu(result) or dup
{
    const int lane  = threadIdx.x & 31;
    const int wpair = blockIdx.x * (blockDim.x >> 5) + (threadIdx.x >> 5);
    if (wpair >= N_NODES / 32) return;   // wave-uniform: EXEC stays all-1s

    const int m     = lane & 15;   // M row within tile (A), N col within tile (B/C)
    const int khalf = lane >> 4;   // lanes 16-31 carry K+2,K+3 (A) / rows K+2,K+3 (B)
    const int row0  = wpair * 32;        // M-tile 0: rows row0..row0+15
    const int rowA0 = row0 + m;
    const int rowA1 = row0 + 16 + m;     // M-tile 1: rows row0+16..row0+31

    constexpr int NT = NCOLS / 16;
    v8f acc0[NT], acc1[NT];
#pragma unroll
    for (int i = 0; i < NT; ++i)
#pragma unroll
        for (int j = 0; j < 8; ++j) { acc0[i][j] = 0.0f; acc1[i][j] = 0.0f; }

    for (int k0 = 0; k0 < 128; k0 += 4) {
        const int kk = k0 + 2 * khalf;
        // A tiles: lane holds A[m][kk], A[m][kk+1]  (64-bit loads)
        v2f aS0 = *(const v2f*)(A_self  + rowA0 * 128 + kk);
        v2f aS1 = *(const v2f*)(A_self  + rowA1 * 128 + kk);
        v2f aN0 = *(const v2f*)(A_neigh + rowA0 * 128 + kk);
        v2f aN1 = *(const v2f*)(A_neigh + rowA1 * 128 + kk);
#pragma unroll
        for (int nt = 0; nt < NT; ++nt) {
            const int n = nt * 16 + m;
            // B tiles: vgpr0 = W[kk][n], vgpr1 = W[kk+1][n] — shared by both M-tiles
            v2f bS, bN;
            bS.x = Wself [(kk    ) * NCOLS + n];
            bS.y = Wself [(kk + 1) * NCOLS + n];
            bN.x = Wneigh[(kk    ) * NCOLS + n];
            bN.y = Wneigh[(kk + 1) * NCOLS + n];
            acc0[nt] = __builtin_amdgcn_wmma_f32_16x16x4_f32(
                false, aS0, false, bS, (short)0, acc0[nt], false, false);
            acc0[nt] = __builtin_amdgcn_wmma_f32_16x16x4_f32(
                false, aN0, false, bN, (short)0, acc0[nt], false, false);
            acc1[nt] = __builtin_amdgcn_wmma_f32_16x16x4_f32(
                false, aS1, false, bS, (short)0, acc1[nt], false, false);
            acc1[nt] = __builtin_amdgcn_wmma_f32_16x16x4_f32(
                false, aN1, false, bN, (short)0, acc1[nt], false, false);
        }
    }

#pragma unroll
    for (int nt = 0; nt < NT; ++nt) {
        const float bv = bias[nt * 16 + m];
#pragma unroll
        for (int r = 0; r < 8; ++r) {
            // C/D layout: VGPR r -> M = r (lanes 0-15), M = r+8 (lanes 16-31); N = m
            const int mm = r + 8 * khalf;
            const float v0 = acc0[nt][r] + bv;
            const float v1 = acc1[nt][r] + bv;
            const int idx0 = (row0 + mm) * NCOLS + nt * 16 + m;
            const int idx1 = (row0 + 16 + mm) * NCOLS + nt * 16 + m;
            out0[idx0] = v0;
            out0[idx1] = v1;
            out1[idx0] = RELU1 ? fmaxf(v0, 0.0f) : v0;
            out1[idx1] = RELU1 ? fmaxf(v1, 0.0f) : v1;
        }
    }
}

extern "C" void kernel_launch(void* const* d_in, const int* in_sizes, int n_in,
                              void* d_out, int out_size, void* d_ws, size_t ws_size,
                              hipStream_t stream) {
    const float* x   = (const float*)d_in[0];
    const float* Ws1 = (const float*)d_in[1];
    const float* Wn1 = (const float*)d_in[2];
    const float* b1  = (const float*)d_in[3];
    const float* Ws2 = (const float*)d_in[4];
    const float* Wn2 = (const float*)d_in[5];
    const float* b2  = (const float*)d_in[6];
    const int*   src = (const int*)d_in[7];
    const int*   dst = (const int*)d_in[8];

    float* out = (float*)d_out;
    float* h2a = out;                                   // h2 copy 1 [N,64]
    float* h1  = h2a + (size_t)N_NODES * OUT_C;         // h1        [N,128]
    float* h1r = h1  + (size_t)N_NODES * HID;           // relu(h1)  [N,128]
    float* h2b = h1r + (size_t)N_NODES * HID;           // h2 copy 2 [N,64]

    float* agg = (float*)d_ws;                          // [N,128] = 51.2 MB
    float* deg = agg + (size_t)N_NODES * 128;           // [N] -> becomes inv_deg

    // zero agg + deg (contiguous N*129 floats)
    {
        int n = N_NODES * 129;
        zero_f32<<<(n + 255) / 256, 256, 0, stream>>>(agg, n);
    }
    deg_kernel<<<N_EDGES / 256, 256, 0, stream>>>(dst, deg);
    invdeg_kernel<<<(N_NODES + 255) / 256, 256, 0, stream>>>(deg);

    // layer 1: scatter x (pre-scaled by inv_deg[dst]), then fused GEMM
    scatter_kernel<<<N_EDGES / 2, 256, 0, stream>>>(x, src, dst, deg, agg);
    {
        int pairs = N_NODES / 32;                       // 3125, exact
        sage_gemm<HID, true><<<(pairs + 3) / 4, 128, 0, stream>>>(
            x, agg, Ws1, Wn1, b1, h1, h1r);
    }

    // layer 2: re-zero agg, scatter relu(h1), fused GEMM (writes both h2 copies)
    {
        int n = N_NODES * 128;
        zero_f32<<<(n + 255) / 256, 256, 0, stream>>>(agg, n);
    }
    scatter_kernel<<<N_EDGES / 2, 256, 0, stream>>>(h1r, src, dst, deg, agg);
    {
        int pairs = N_NODES / 32;
        sage_gemm<OUT_C, false><<<(pairs + 3) / 4, 128, 0, stream>>>(
            h1r, agg, Ws2, Wn2, b2, h2a, h2b);
    }
}